// TrainingDetectionLayer_23536420782359
// MI455X (gfx1250) — compile-verified
//
#include <hip/hip_runtime.h>

#define B_    16
#define N_    6000
#define G_    256
#define T_    200
#define POSN  66
#define NEGS  134
#define KMAX  134
#define CHUNKS 24                 // ceil(N_/256)
#define TOPK_THREADS 512
#define EPSF  1e-8f
#define NINF  (-__builtin_inff())

typedef __attribute__((ext_vector_type(4))) unsigned int v4u;
typedef __attribute__((ext_vector_type(8))) int          v8i;
typedef __attribute__((ext_vector_type(4))) int          v4i;

// ---------------------------------------------------------------------------
// CDNA5 Tensor Data Mover: DMA a contiguous run of `ndw` dwords from global
// memory into LDS at byte offset `lds_addr`, described as a 1-D tensor tile.
// D# bit layout per CDNA5 ISA.
// ---------------------------------------------------------------------------
__device__ __forceinline__ void tdm_load_dwords_1d(unsigned lds_addr,
                                                   unsigned long long gaddr,
                                                   unsigned ndw) {
#if __has_builtin(__builtin_amdgcn_tensor_load_to_lds)
  v4u g0;
  g0.x = 1u;                                             // count=1, user desc
  g0.y = lds_addr;                                       // LDS byte address
  g0.z = (unsigned)(gaddr & 0xFFFFFFFFull);              // global_addr[31:0]
  g0.w = (unsigned)((gaddr >> 32) & 0x1FFFFFFull)        // global_addr[56:32]
       | (2u << 30);                                     // type=2 ("image")
  v8i g1;
  g1[0] = 0x00020000;                                    // data_size=4B
  g1[1] = (int)((ndw & 0xFFFFu) << 16);                  // tensor_dim0[15:0]
  g1[2] = (int)(((ndw >> 16) & 0xFFFFu) | (1u << 16));   // dim0 hi | tensor_dim1=1
  g1[3] = (int)((ndw & 0xFFFFu) << 16);                  // tile_dim0
  g1[4] = 0;                                             // tile_dim1/2 unused
  g1[5] = (int)ndw;                                      // tensor_dim0_stride
  g1[6] = 0;
  g1[7] = 0;
  v4i z4 = {0, 0, 0, 0};
#if defined(__clang_major__) && (__clang_major__ >= 23)
  v8i z8 = {0, 0, 0, 0, 0, 0, 0, 0};
  __builtin_amdgcn_tensor_load_to_lds(g0, g1, z4, z4, z8, 0);
#else
  __builtin_amdgcn_tensor_load_to_lds(g0, g1, z4, z4, 0);
#endif
#else
  (void)lds_addr; (void)gaddr; (void)ndw;
#endif
}

__device__ __forceinline__ void tensor_wait0() {
#if __has_builtin(__builtin_amdgcn_s_wait_tensorcnt)
  __builtin_amdgcn_s_wait_tensorcnt(0);
#else
  asm volatile("s_wait_tensorcnt 0x0" ::: "memory");
#endif
}

// ---------------------------------------------------------------------------
// Kernel 1: per-ROI IoU reductions against GT staged in LDS via TDM.
// ---------------------------------------------------------------------------
__global__ __launch_bounds__(256) void tdl_iou_kernel(
    const float* __restrict__ rois, const int* __restrict__ gt_ids,
    const float* __restrict__ gt_boxes, const float* __restrict__ pos_score,
    const float* __restrict__ neg_score, float* __restrict__ pos_key,
    float* __restrict__ neg_key, int* __restrict__ assign) {
  __shared__ float4 s_gt[G_];
  __shared__ int s_ids[G_];

  const int tid = threadIdx.x;
  const int b = blockIdx.x / CHUNKS;
  const int chunk = blockIdx.x % CHUNKS;
  const int r = chunk * 256 + tid;

  if (tid < G_) s_ids[tid] = gt_ids[(size_t)b * G_ + tid];

#if __has_builtin(__builtin_amdgcn_tensor_load_to_lds)
  if (tid == 0) {
    unsigned lds = (unsigned)(unsigned long long)(const void*)&s_gt[0];
    unsigned long long ga =
        (unsigned long long)(const void*)(gt_boxes + (size_t)b * G_ * 4);
    tdm_load_dwords_1d(lds, ga, G_ * 4);   // 4KB tile -> LDS via TDM
  }
  tensor_wait0();
#else
  if (tid < G_) s_gt[tid] = ((const float4*)gt_boxes)[(size_t)b * G_ + tid];
#endif
  __syncthreads();

  if (r < N_) {
    const size_t o = (size_t)b * N_ + r;
    __builtin_prefetch(pos_score + o, 0, 1);   // global_prefetch_b8
    __builtin_prefetch(neg_score + o, 0, 1);

    const float4 rb = *(const float4*)(rois + o * 4);
    const bool roi_valid =
        (rb.x != 0.f) || (rb.y != 0.f) || (rb.z != 0.f) || (rb.w != 0.f);
    const float area_r = (rb.z - rb.x) * (rb.w - rb.y);

    float nc_max = NINF;
    int nc_arg = 0;
    float cr_max = 0.f;
    for (int g = 0; g < G_; ++g) {
      const float4 gb = s_gt[g];                // ds b128
      const int id = s_ids[g];
      const bool gv =
          (gb.x != 0.f) || (gb.y != 0.f) || (gb.z != 0.f) || (gb.w != 0.f);
      const float y1 = fmaxf(rb.x, gb.x), x1 = fmaxf(rb.y, gb.y);
      const float y2 = fminf(rb.z, gb.z), x2 = fminf(rb.w, gb.w);
      const float inter = fmaxf(y2 - y1, 0.f) * fmaxf(x2 - x1, 0.f);
      const float area_g = (gb.z - gb.x) * (gb.w - gb.y);
      const float iou = inter / fmaxf(area_r + area_g - inter, EPSF);
      const float vnc = (gv && id > 0) ? iou : -1.0f;
      if (vnc > nc_max) { nc_max = vnc; nc_arg = g; }   // first-argmax
      const float vcr = (gv && id < 0) ? iou : 0.0f;
      cr_max = fmaxf(cr_max, vcr);
    }
    const bool pos = roi_valid && (nc_max >= 0.5f);
    const bool neg = roi_valid && (nc_max < 0.5f) && (cr_max < 0.001f);
    pos_key[o] = pos ? pos_score[o] : NINF;
    neg_key[o] = neg ? neg_score[o] : NINF;
    assign[o] = nc_arg;
  }
}

// ---------------------------------------------------------------------------
// Kernel 2: exact top-K radix select; one block per (image, pos|neg).
// Produces sel[0..K-1] = indices in (key desc, index asc) order — exactly the
// first K entries of a stable argsort on descending keys — plus the count of
// keys != -inf. All heavy state lives in LDS (24KB keys, 1KB bins).
// ---------------------------------------------------------------------------
__device__ __forceinline__ unsigned f32_ord(float v) {
  unsigned u = __float_as_uint(v);
  return (u & 0x80000000u) ? ~u : (u | 0x80000000u);
}

__global__ __launch_bounds__(TOPK_THREADS) void tdl_topk_kernel(
    const float* __restrict__ pos_key, const float* __restrict__ neg_key,
    int* __restrict__ sel_ws, int* __restrict__ counts) {
  __shared__ unsigned ukeys[N_];
  __shared__ int bins[256];
  __shared__ int misc[8];     // [0]=valid, [1]=bin, [2]=above-in-pass,
                              // [3]=above list cnt, [4]=eq list cnt
  __shared__ unsigned ab_ou[KMAX];
  __shared__ int ab_idx[KMAX];
  __shared__ int eq_idx[256];

  const int tid = threadIdx.x;
  const int b = blockIdx.x >> 1;
  const int which = blockIdx.x & 1;          // 0 = positives, 1 = negatives
  const int K = which ? NEGS : POSN;
  const float* __restrict__ gkeys =
      (which ? neg_key : pos_key) + (size_t)b * N_;
  int* __restrict__ sel_out = sel_ws + (size_t)b * T_ + (which ? POSN : 0);

  // ---- load keys, convert to order-preserving u32, count valid ----
  if (tid == 0) misc[0] = 0;
  __syncthreads();
  int c = 0;
  for (int i = tid; i < N_; i += TOPK_THREADS) {
    const float v = gkeys[i];
    ukeys[i] = f32_ord(v);
    if (v != NINF) c++;
  }
  atomicAdd(&misc[0], c);
  __syncthreads();
  const int valid = misc[0];

  // ---- 4x 8-bit radix passes to find the K-th largest ordered key ----
  unsigned prefix = 0;
  int remaining = K;
  for (int p = 3; p >= 0; --p) {
    if (tid < 256) bins[tid] = 0;
    __syncthreads();
    const unsigned shift = (unsigned)p * 8u;
    const unsigned himask = (p == 3) ? 0u : (0xFFFFFFFFu << (shift + 8u));
    for (int i = tid; i < N_; i += TOPK_THREADS) {
      const unsigned ou = ukeys[i];
      if ((ou & himask) == prefix) atomicAdd(&bins[(ou >> shift) & 255u], 1);
    }
    __syncthreads();
    if (tid == 0) {
      int acc = 0, bsel = 0;
      for (int v = 255; v >= 0; --v) {
        const int nb = bins[v];
        if (acc + nb >= remaining) { bsel = v; break; }
        acc += nb;
      }
      misc[1] = bsel;
      misc[2] = acc;          // strictly-above count within this prefix
    }
    __syncthreads();
    prefix |= ((unsigned)misc[1]) << shift;
    remaining -= misc[2];
    __syncthreads();
  }
  const unsigned Tou = prefix;   // exact K-th largest ordered key
  const int rem = remaining;     // how many == Tou entries to take

  // ---- compact (> Tou) and (== Tou) candidate sets ----
  if (tid == 0) { misc[3] = 0; misc[4] = 0; }
  __syncthreads();
  for (int i = tid; i < N_; i += TOPK_THREADS) {
    const unsigned ou = ukeys[i];
    if (ou > Tou) {
      const int p = atomicAdd(&misc[3], 1);
      if (p < KMAX) { ab_ou[p] = ou; ab_idx[p] = i; }
    } else if (ou == Tou) {
      const int p = atomicAdd(&misc[4], 1);
      if (p < 256) eq_idx[p] = i;
    }
  }
  __syncthreads();
  int above_cnt = misc[3];
  if (above_cnt > KMAX) above_cnt = KMAX;      // invariant: above_cnt < K
  int eq_stored = misc[4];
  if (eq_stored > 256) eq_stored = 256;

  // ---- rank-place: slots 0..above-1 sorted (key desc, idx asc) ----
  if (tid < above_cnt) {
    const unsigned mou = ab_ou[tid];
    const int midx = ab_idx[tid];
    int rank = 0;
    for (int j = 0; j < above_cnt; ++j) {
      const unsigned oj = ab_ou[j];
      const int ij = ab_idx[j];
      if (oj > mou || (oj == mou && ij < midx)) rank++;
    }
    sel_out[rank] = midx;
  }
  // ---- slots above..K-1: lowest-index entries equal to the threshold ----
  if (tid < eq_stored) {
    const int midx = eq_idx[tid];
    int rank = 0;
    for (int j = 0; j < eq_stored; ++j)
      if (eq_idx[j] < midx) rank++;
    if (rank < rem) sel_out[above_cnt + rank] = midx;
  }
  if (tid == 0) counts[b * 2 + which] = valid;
}

// ---------------------------------------------------------------------------
// Kernel 3: gather + bbox-offset + emit (one block per image).
// Output layout (floats): rois [B,T,4] | ids [B,T] | off [B,T,4].
// ---------------------------------------------------------------------------
__global__ __launch_bounds__(256) void tdl_emit_kernel(
    const float* __restrict__ rois, const int* __restrict__ gt_ids,
    const float* __restrict__ gt_boxes, const int* __restrict__ assign,
    const int* __restrict__ sel_ws, const int* __restrict__ counts,
    float* __restrict__ out) {
  const int tid = threadIdx.x;
  const int b = blockIdx.x;
  if (tid >= T_) return;

  const int pos_total = counts[b * 2 + 0];
  const int neg_total = counts[b * 2 + 1];
  const int pos_count = pos_total < POSN ? pos_total : POSN;
  // neg_target = int32(float32(1/0.33) * pos_count) - pos_count  (truncation)
  int neg_count = (int)(((float)(1.0 / 0.33)) * (float)pos_count) - pos_count;
  if (neg_count > NEGS) neg_count = NEGS;
  if (neg_count > neg_total) neg_count = neg_total;

  const float* roib = rois + (size_t)b * N_ * 4;
  const float* gtbb = gt_boxes + (size_t)b * G_ * 4;
  const int* gidb = gt_ids + (size_t)b * G_;
  const int* asgb = assign + (size_t)b * N_;
  float* out_rois = out;                          // [B,T,4]
  float* out_ids = out + (size_t)B_ * T_ * 4;     // [B,T]
  float* out_off = out + (size_t)B_ * T_ * 5;     // [B,T,4]

  const int idx = sel_ws[(size_t)b * T_ + tid];
  const float4 rb = *(const float4*)(roib + (size_t)idx * 4);
  const size_t t = (size_t)b * T_ + tid;

  if (tid < POSN) {
    const bool valid = tid < pos_count;
    const int a = asgb[idx];
    const float4 gb = *(const float4*)(gtbb + (size_t)a * 4);
    const int gid = gidb[a];

    const float h = fmaxf(rb.z - rb.x, EPSF), w = fmaxf(rb.w - rb.y, EPSF);
    const float cy = rb.x + 0.5f * h, cx = rb.y + 0.5f * w;
    const float gh = fmaxf(gb.z - gb.x, EPSF), gw = fmaxf(gb.w - gb.y, EPSF);
    const float gcy = gb.x + 0.5f * gh, gcx = gb.y + 0.5f * gw;
    const float o0 = ((gcy - cy) / h) / 0.1f;
    const float o1 = ((gcx - cx) / w) / 0.1f;
    const float o2 = logf(gh / h) / 0.2f;
    const float o3 = logf(gw / w) / 0.2f;

    out_rois[t * 4 + 0] = valid ? rb.x : 0.f;
    out_rois[t * 4 + 1] = valid ? rb.y : 0.f;
    out_rois[t * 4 + 2] = valid ? rb.z : 0.f;
    out_rois[t * 4 + 3] = valid ? rb.w : 0.f;
    out_ids[t] = valid ? (float)gid : 0.f;
    out_off[t * 4 + 0] = valid ? o0 : 0.f;
    out_off[t * 4 + 1] = valid ? o1 : 0.f;
    out_off[t * 4 + 2] = valid ? o2 : 0.f;
    out_off[t * 4 + 3] = valid ? o3 : 0.f;
  } else {
    const bool valid = (tid - POSN) < neg_count;
    out_rois[t * 4 + 0] = valid ? rb.x : 0.f;
    out_rois[t * 4 + 1] = valid ? rb.y : 0.f;
    out_rois[t * 4 + 2] = valid ? rb.z : 0.f;
    out_rois[t * 4 + 3] = valid ? rb.w : 0.f;
    out_ids[t] = 0.f;
    out_off[t * 4 + 0] = 0.f;
    out_off[t * 4 + 1] = 0.f;
    out_off[t * 4 + 2] = 0.f;
    out_off[t * 4 + 3] = 0.f;
  }
}

// ---------------------------------------------------------------------------
extern "C" void kernel_launch(void* const* d_in, const int* in_sizes, int n_in,
                              void* d_out, int out_size, void* d_ws,
                              size_t ws_size, hipStream_t stream) {
  (void)in_sizes; (void)n_in; (void)out_size; (void)ws_size;
  const float* rois = (const float*)d_in[0];
  const int* gt_ids = (const int*)d_in[1];
  const float* gt_boxes = (const float*)d_in[2];
  const float* pos_score = (const float*)d_in[3];
  const float* neg_score = (const float*)d_in[4];
  float* out = (float*)d_out;

  float* pos_key = (float*)d_ws;                       // B*N floats
  float* neg_key = pos_key + (size_t)B_ * N_;          // B*N floats
  int* assign = (int*)(neg_key + (size_t)B_ * N_);     // B*N ints
  int* sel_ws = assign + (size_t)B_ * N_;              // B*T ints
  int* counts = sel_ws + (size_t)B_ * T_;              // B*2 ints

  tdl_iou_kernel<<<dim3(B_ * CHUNKS), dim3(256), 0, stream>>>(
      rois, gt_ids, gt_boxes, pos_score, neg_score, pos_key, neg_key, assign);
  tdl_topk_kernel<<<dim3(B_ * 2), dim3(TOPK_THREADS), 0, stream>>>(
      pos_key, neg_key, sel_ws, counts);
  tdl_emit_kernel<<<dim3(B_), dim3(256), 0, stream>>>(
      rois, gt_ids, gt_boxes, assign, sel_ws, counts, out);
}